// HamNetEncoder_50749333569599
// MI455X (gfx1250) — compile-verified
//
#include <hip/hip_runtime.h>
#include <hip/hip_bf16.h>
#include <math.h>

#define N_NODES 20000
#define N_EDGES 320000
#define HID 256
#define LN_EPS 1e-5f

// GEMM tiling: 16-row strip per block, K staged in 32-row chunks of B in LDS
#define KC 32
#define LDS_STRIDE 264   // 256 + 8 pad: K-half lane groups hit disjoint banks

typedef __attribute__((ext_vector_type(2))) float v2f;
typedef __attribute__((ext_vector_type(8))) float v8f;

// ---- monotone float<->uint key for atomic max over signed floats ----
__device__ __forceinline__ unsigned f2key(float f) {
  unsigned u = __float_as_uint(f);
  return (u & 0x80000000u) ? ~u : (u | 0x80000000u);
}
__device__ __forceinline__ float key2f(unsigned k) {
  unsigned u = (k & 0x80000000u) ? (k & 0x7fffffffu) : ~k;
  return __uint_as_float(u);
}

__device__ __forceinline__ float waveReduceSum(float v) {
  #pragma unroll
  for (int off = 16; off; off >>= 1) v += __shfl_xor(v, off, 32);
  return v;
}
__device__ __forceinline__ float waveReduceMax(float v) {
  #pragma unroll
  for (int off = 16; off; off >>= 1) v = fmaxf(v, __shfl_xor(v, off, 32));
  return v;
}

// ---------------- utility fills ----------------
__global__ void k_fill_f32(float* p, int n, float v) {
  int i = blockIdx.x * blockDim.x + threadIdx.x;
  if (i < n) p[i] = v;
}
__global__ void k_fill_u32(unsigned* p, int n, unsigned v) {
  int i = blockIdx.x * blockDim.x + threadIdx.x;
  if (i < n) p[i] = v;
}

// ---------------- embedding gather ----------------
__global__ void k_gather(const float* __restrict__ emb, const int* __restrict__ ids,
                         float* __restrict__ h) {
  int i = blockIdx.x * 256 + threadIdx.x;   // grid sized exactly N*H/256
  int n = i >> 8, c = i & 255;
  h[i] = emb[ids[n] * HID + c];
}

// ---------------- fp32 WMMA GEMM: C[M,256] = A[M,256] * B[256,256] ----------------
// block = 128 threads (4 waves); block owns a 16-row strip; wave w owns cols [64w,64w+64)
// B is staged in LDS via GLOBAL_LOAD_ASYNC_TO_LDS_B128 (ASYNCcnt), double-buffered:
// copy of K-chunk i+1 overlaps WMMA compute on chunk i.
// V_WMMA_F32_16X16X4_F32 fragments (ISA 7.12.2):
//   A 16x4 f32 : lane L -> row m=L%16 ; VGPR v -> K = 2*(L/16)+v
//   B 4x16 f32 : lane L -> col n=L%16 ; VGPR v -> K = 2*(L/16)+v
//   C 16x16    : lane L -> col n=L%16 ; VGPR r -> row = r + 8*(L/16)
__global__ __launch_bounds__(128)
void k_gemm_wmma(const float* __restrict__ A, const float* __restrict__ B,
                 float* __restrict__ C) {
  __shared__ float sB[2][KC * LDS_STRIDE];

  const int tid  = threadIdx.x;
  const int lane = tid & 31;
  const int wv   = tid >> 5;
  const int row0 = blockIdx.x << 4;
  const int col0 = wv << 6;
  const int m    = lane & 15;
  const int kh   = lane >> 4;        // 0 or 1: K-half selector

  // per-thread async-copy share of one 32x256 B chunk: 16 x b128 (fully coalesced)
  auto issue_chunk = [&](int kc0, int buf) {
    #pragma unroll
    for (int l = 0; l < 16; ++l) {
      int j  = tid + (l << 7);                 // 0..2047 (b128 index in chunk)
      int r  = j >> 6;                         // chunk row 0..31
      int c4 = (j & 63) << 2;                  // column (float index)
      unsigned lds   = (unsigned)(size_t)&sB[buf][r * LDS_STRIDE + c4];
      const float* g = B + kc0 * HID + (j << 2);
      asm volatile("global_load_async_to_lds_b128 %0, %1, off"
                   :: "v"(lds), "v"(g) : "memory");
    }
  };

  issue_chunk(0, 0);

  v8f acc0 = {}, acc1 = {}, acc2 = {}, acc3 = {};
  const float* Arow = A + (row0 + m) * HID + 2 * kh;

  for (int ch = 0; ch < HID / KC; ++ch) {
    if (ch < HID / KC - 1) {
      issue_chunk((ch + 1) * KC, (ch + 1) & 1);
      asm volatile("s_wait_asynccnt 0x10" ::: "memory");  // chunk ch landed; 16 newest in flight
    } else {
      asm volatile("s_wait_asynccnt 0x0" ::: "memory");
    }
    __syncthreads();

    __builtin_prefetch(Arow + ch * KC + KC, 0, 3);        // near-scope prefetch of next A chunk
    const int buf = ch & 1;
    #pragma unroll
    for (int kb = 0; kb < KC; kb += 4) {
      v2f a = *(const v2f*)(Arow + ch * KC + kb);         // A[m][.. + 2kh .. +1]
      const int bi = (kb + 2 * kh) * LDS_STRIDE + col0 + m;
      v2f b0, b1, b2, b3;
      b0.x = sB[buf][bi +  0];             b0.y = sB[buf][bi + LDS_STRIDE +  0];
      b1.x = sB[buf][bi + 16];             b1.y = sB[buf][bi + LDS_STRIDE + 16];
      b2.x = sB[buf][bi + 32];             b2.y = sB[buf][bi + LDS_STRIDE + 32];
      b3.x = sB[buf][bi + 48];             b3.y = sB[buf][bi + LDS_STRIDE + 48];
      acc0 = __builtin_amdgcn_wmma_f32_16x16x4_f32(false, a, false, b0, (short)0, acc0, false, false);
      acc1 = __builtin_amdgcn_wmma_f32_16x16x4_f32(false, a, false, b1, (short)0, acc1, false, false);
      acc2 = __builtin_amdgcn_wmma_f32_16x16x4_f32(false, a, false, b2, (short)0, acc2, false, false);
      acc3 = __builtin_amdgcn_wmma_f32_16x16x4_f32(false, a, false, b3, (short)0, acc3, false, false);
    }
    __syncthreads();
  }

  float* Crow = C + (row0 + 8 * kh) * HID + col0 + m;
  #pragma unroll
  for (int r = 0; r < 8; ++r) {
    Crow[r * HID +  0] = acc0[r];
    Crow[r * HID + 16] = acc1[r];
    Crow[r * HID + 32] = acc2[r];
    Crow[r * HID + 48] = acc3[r];
  }
}

// ---------------- s_src / s_dst matrix-vector (wave per node) ----------------
__global__ __launch_bounds__(256)
void k_sv(const float* __restrict__ P, const float* __restrict__ we,
          float* __restrict__ ssrc, float* __restrict__ sdst) {
  int lane = threadIdx.x & 31;
  int node = (blockIdx.x << 3) + (threadIdx.x >> 5);
  const float* row = P + node * HID;
  float s0 = 0.f, s1 = 0.f;
  #pragma unroll
  for (int i = 0; i < 8; ++i) {
    int c = lane + (i << 5);
    float h = row[c];
    s0 += h * we[c];
    s1 += h * we[HID + c];
  }
  s0 = waveReduceSum(s0);
  s1 = waveReduceSum(s1);
  if (!lane) { ssrc[node] = s0; sdst[node] = s1; }
}

// ---------------- edge score + global max ----------------
__global__ void k_edge_score(const float* __restrict__ ssrc, const float* __restrict__ sdst,
                             const int* __restrict__ src, const int* __restrict__ dst,
                             float* __restrict__ scores, unsigned* maxkey, int E) {
  int e = blockIdx.x * blockDim.x + threadIdx.x;
  float sc = -INFINITY;
  if (e < E) {
    float v = ssrc[src[e]] + sdst[dst[e]];
    sc = (v >= 0.f) ? v : 0.2f * v;          // leaky_relu(0.2)
    scores[e] = sc;
  }
  sc = waveReduceMax(sc);
  if (!(threadIdx.x & 31)) atomicMax(maxkey, f2key(sc));
}

// ---------------- attn = exp(score - max); denom segment-sum ----------------
__global__ void k_edge_attn(const float* __restrict__ scores, const unsigned* __restrict__ maxkey,
                            const int* __restrict__ dst, float* __restrict__ attn,
                            float* __restrict__ den, int E) {
  int e = blockIdx.x * blockDim.x + threadIdx.x;
  if (e >= E) return;
  float mx = key2f(*maxkey);
  float a = expf(scores[e] - mx);
  attn[e] = a;
  atomicAdd(&den[dst[e]], a);
}

// ---------------- agg[dst] += h[src] * attn (float4 per thread) ----------------
__global__ __launch_bounds__(256)
void k_edge_agg(const float* __restrict__ P, const float* __restrict__ attn,
                const int* __restrict__ src, const int* __restrict__ dst,
                float* __restrict__ agg) {
  int i = blockIdx.x * blockDim.x + threadIdx.x;   // grid = E*64 threads exactly
  int e = i >> 6, q = (i & 63) << 2;
  float a = attn[e];
  float4 hv = *(const float4*)(P + src[e] * HID + q);
  float* o = agg + dst[e] * HID + q;
  atomicAdd(o + 0, hv.x * a);
  atomicAdd(o + 1, hv.y * a);
  atomicAdd(o + 2, hv.z * a);
  atomicAdd(o + 3, hv.w * a);
}

// ---------------- residual + LayerNorm + exact GELU (wave per node, in place) ----------------
__global__ __launch_bounds__(256)
void k_finalize(const float* __restrict__ agg, const float* __restrict__ den,
                const float* __restrict__ g, const float* __restrict__ b,
                float* __restrict__ h) {
  int lane = threadIdx.x & 31;
  int node = (blockIdx.x << 3) + (threadIdx.x >> 5);
  float inv = 1.0f / (den[node] + 1e-6f);
  float x[8];
  float s = 0.f;
  #pragma unroll
  for (int i = 0; i < 8; ++i) {
    int c = lane + (i << 5);
    x[i] = agg[node * HID + c] * inv + h[node * HID + c];
    s += x[i];
  }
  s = waveReduceSum(s);
  float mu = s * (1.0f / HID);
  float vs = 0.f;
  #pragma unroll
  for (int i = 0; i < 8; ++i) { float d = x[i] - mu; vs += d * d; }
  vs = waveReduceSum(vs);
  float rstd = rsqrtf(vs * (1.0f / HID) + LN_EPS);
  #pragma unroll
  for (int i = 0; i < 8; ++i) {
    int c = lane + (i << 5);
    float y = (x[i] - mu) * rstd * g[c] + b[c];
    y = 0.5f * y * (1.0f + erff(y * 0.70710678118654752f));   // exact GELU
    h[node * HID + c] = y;
  }
}

// ---------------- pooling: s[n] = sum_j tanh(proj+b)*w ; masked; global max ----------------
__global__ __launch_bounds__(256)
void k_pool_s(const float* __restrict__ P, const float* __restrict__ pb,
              const float* __restrict__ pw, const int* __restrict__ ids,
              float* __restrict__ spool, unsigned* maxkey) {
  int lane = threadIdx.x & 31;
  int node = (blockIdx.x << 3) + (threadIdx.x >> 5);
  const float* row = P + node * HID;
  float s = 0.f;
  #pragma unroll
  for (int i = 0; i < 8; ++i) {
    int c = lane + (i << 5);
    s += tanhf(row[c] + pb[c]) * pw[c];
  }
  s = waveReduceSum(s);
  if (!lane) {
    float sv = (ids[node] != 0) ? s : -INFINITY;
    spool[node] = sv;
    atomicMax(maxkey, f2key(sv));
  }
}

__global__ void k_pool_exp(const float* __restrict__ spool, const unsigned* __restrict__ maxkey,
                           float* __restrict__ wraw, float* __restrict__ sum, int n) {
  int i = blockIdx.x * blockDim.x + threadIdx.x;
  float e = 0.f;
  if (i < n) {
    float mx = key2f(*maxkey);
    float sv = spool[i];
    e = (sv == -INFINITY) ? 0.f : expf(sv - mx);   // masked nodes -> 0 (nan_to_num)
    wraw[i] = e;
  }
  e = waveReduceSum(e);
  if (!(threadIdx.x & 31)) atomicAdd(sum, e);
}

__global__ void k_pool_norm(const float* __restrict__ wraw, const float* __restrict__ sum,
                            float* __restrict__ wts, int n) {
  int i = blockIdx.x * blockDim.x + threadIdx.x;
  if (i >= n) return;
  float S = *sum;
  wts[i] = (S > 0.f) ? wraw[i] / S : 0.f;
}

// ---------------- ctx[c] = sum_n wts[n] * h[n][c] ----------------
__global__ __launch_bounds__(256)
void k_ctx(const float* __restrict__ wts, const float* __restrict__ h,
           float* __restrict__ ctx) {
  int c = threadIdx.x;
  float acc = 0.f;
  for (int n = blockIdx.x; n < N_NODES; n += gridDim.x)
    acc += wts[n] * h[n * HID + c];
  atomicAdd(&ctx[c], acc);
}

extern "C" void kernel_launch(void* const* d_in, const int* in_sizes, int n_in,
                              void* d_out, int out_size, void* d_ws, size_t ws_size,
                              hipStream_t stream) {
  const int*   ids = (const int*)d_in[0];
  const int*   ei  = (const int*)d_in[1];
  const float* emb = (const float*)d_in[2];
  const float* W0  = (const float*)d_in[3];
  const float* we0 = (const float*)d_in[4];
  const float* g0  = (const float*)d_in[5];
  const float* b0  = (const float*)d_in[6];
  const float* W1  = (const float*)d_in[7];
  const float* we1 = (const float*)d_in[8];
  const float* g1  = (const float*)d_in[9];
  const float* b1  = (const float*)d_in[10];
  const float* pW  = (const float*)d_in[11];
  const float* pb  = (const float*)d_in[12];
  const float* pw  = (const float*)d_in[13];
  const int* src = ei;
  const int* dst = ei + N_EDGES;
  float* out = (float*)d_out;            // [0,256): ctx ; [256,20256): wts

  float* ws = (float*)d_ws;
  size_t off = 0;
  float* H   = ws + off; off += (size_t)N_NODES * HID;   // node features (in-place per layer)
  float* P   = ws + off; off += (size_t)N_NODES * HID;   // projected features / pool proj
  float* G   = ws + off; off += (size_t)N_NODES * HID;   // aggregation accumulator
  float* SS  = ws + off; off += N_NODES;                 // s_src
  float* SD  = ws + off; off += N_NODES;                 // s_dst
  float* DEN = ws + off; off += N_NODES;                 // softmax denominators
  float* SCR = ws + off; off += N_EDGES;                 // edge scores
  float* ATT = ws + off; off += N_EDGES;                 // edge exp weights
  float* SPL = ws + off; off += N_NODES;                 // pool scores
  float* WRW = ws + off; off += N_NODES;                 // unnormalized pool weights
  unsigned* MAXK = (unsigned*)(ws + off); off += 4;      // max-reduction key
  float* SUM = ws + off; off += 4;                       // pool softmax sum

  const int NH = N_NODES * HID;

  k_gather<<<NH / 256, 256, 0, stream>>>(emb, ids, H);

  for (int layer = 0; layer < 2; ++layer) {
    const float* W  = layer ? W1  : W0;
    const float* we = layer ? we1 : we0;
    const float* g  = layer ? g1  : g0;
    const float* b  = layer ? b1  : b0;

    k_fill_f32<<<NH / 256, 256, 0, stream>>>(G, NH, 0.f);
    k_fill_f32<<<(N_NODES + 255) / 256, 256, 0, stream>>>(DEN, N_NODES, 0.f);
    k_fill_u32<<<1, 32, 0, stream>>>(MAXK, 1, 0u);

    k_gemm_wmma<<<N_NODES / 16, 128, 0, stream>>>(H, W, P);
    k_sv<<<N_NODES / 8, 256, 0, stream>>>(P, we, SS, SD);
    k_edge_score<<<(N_EDGES + 255) / 256, 256, 0, stream>>>(SS, SD, src, dst, SCR, MAXK, N_EDGES);
    k_edge_attn<<<(N_EDGES + 255) / 256, 256, 0, stream>>>(SCR, MAXK, dst, ATT, DEN, N_EDGES);
    k_edge_agg<<<(N_EDGES * 64) / 256, 256, 0, stream>>>(P, ATT, src, dst, G);
    k_finalize<<<N_NODES / 8, 256, 0, stream>>>(G, DEN, g, b, H);
  }

  // ---- attention pooling ----
  k_gemm_wmma<<<N_NODES / 16, 128, 0, stream>>>(H, pW, P);
  k_fill_u32<<<1, 32, 0, stream>>>(MAXK, 1, 0u);
  k_fill_f32<<<1, 32, 0, stream>>>(SUM, 1, 0.f);
  k_fill_f32<<<1, 256, 0, stream>>>(out, HID, 0.f);
  k_pool_s<<<N_NODES / 8, 256, 0, stream>>>(P, pb, pw, ids, SPL, MAXK);
  k_pool_exp<<<(N_NODES + 255) / 256, 256, 0, stream>>>(SPL, MAXK, WRW, SUM, N_NODES);
  k_pool_norm<<<(N_NODES + 255) / 256, 256, 0, stream>>>(WRW, SUM, out + HID, N_NODES);
  k_ctx<<<125, 256, 0, stream>>>(out + HID, H, out);
}